// GETLayer_80857054315096
// MI455X (gfx1250) — compile-verified
//
#include <hip/hip_runtime.h>
#include <math.h>

// Problem constants (from reference)
#define NN 50000
#define DD 128
#define KM 32
#define EE2 400000
#define EE3 100000
#define SCALE 0.08838834764831845f   // 1/sqrt(128)
#define BETA_MAXC 5.0f
#define GRAD_CLIPC 1.0f
#define STATE_CLIPC 10.0f
#define DAMPC 0.9999f
#define LN_EPSC 1e-5f

typedef __attribute__((ext_vector_type(2))) float v2f;
typedef __attribute__((ext_vector_type(8))) float v8f;

// ---------- wave32 helpers ----------
__device__ __forceinline__ float wave_sum(float v) {
#pragma unroll
  for (int o = 16; o > 0; o >>= 1) v += __shfl_xor(v, o, 32);
  return v;
}
__device__ __forceinline__ float wave_max(float v) {
#pragma unroll
  for (int o = 16; o > 0; o >>= 1) v = fmaxf(v, __shfl_xor(v, o, 32));
  return v;
}
// monotone float<->uint encoding for atomicMax on floats (handles negatives)
__device__ __forceinline__ unsigned enc_f(float f) {
  unsigned u = __float_as_uint(f);
  return (u & 0x80000000u) ? ~u : (u | 0x80000000u);
}
__device__ __forceinline__ float dec_f(unsigned e) {
  return __uint_as_float((e & 0x80000000u) ? (e & 0x7fffffffu) : ~e);
}
__device__ __forceinline__ void atomAddF(float* p, float v) {
  unsafeAtomicAdd(p, v);  // native global_atomic_add_f32
}
__device__ __forceinline__ float softplusf(float x) {
  return (x > 20.f) ? x : log1pf(expf(x));
}

// ---------- K0: scalars + accumulator init ----------
__global__ void k_scalars(const float* l2r, const float* l3r, const float* lmr,
                          const float* b2r, const float* b3r, const float* bmr,
                          const float* ss, float* sc, float* acc) {
  if (threadIdx.x == 0 && blockIdx.x == 0) {
    sc[0] = softplusf(l2r[0]);
    sc[1] = softplusf(l3r[0]);
    sc[2] = softplusf(lmr[0]);
    sc[3] = fminf(softplusf(b2r[0]), BETA_MAXC);
    sc[4] = fminf(softplusf(b3r[0]), BETA_MAXC);
    sc[5] = fminf(softplusf(bmr[0]), BETA_MAXC);
    sc[6] = ss[0];
    acc[0] = 0.f; acc[1] = 0.f; acc[2] = 0.f;
  }
}

// ---------- K1: segment max/sum init ----------
__global__ void k_seg_init(unsigned* m2e, float* z2, unsigned* m3e, float* z3) {
  int i = blockIdx.x * blockDim.x + threadIdx.x;
  if (i < NN) {
    unsigned ninf = enc_f(-INFINITY);
    m2e[i] = ninf; z2[i] = 0.f;
    m3e[i] = ninf; z3[i] = 0.f;
  }
}

// ---------- K2: LayerNorm stats (one wave per row) ----------
__global__ void k_ln_stats(const float* __restrict__ X, float* mu, float* inv) {
  int lane = threadIdx.x & 31;
  int r = blockIdx.x * 8 + (threadIdx.x >> 5);
  if (r >= NN) return;
  float x0 = X[r * DD + lane], x1 = X[r * DD + lane + 32],
        x2 = X[r * DD + lane + 64], x3 = X[r * DD + lane + 96];
  float m = wave_sum(x0 + x1 + x2 + x3) * (1.f / DD);
  float d0 = x0 - m, d1 = x1 - m, d2 = x2 - m, d3 = x3 - m;
  float v = wave_sum(d0 * d0 + d1 * d1 + d2 * d2 + d3 * d3) * (1.f / DD);
  if (lane == 0) { mu[r] = m; inv[r] = rsqrtf(v + LN_EPSC); }
}

// ---------- K3: pack Wcat (128x640) and WcatT (640x128) ----------
__global__ void k_pack_w(const float* Wq2, const float* Wk2, const float* Wq3,
                         const float* Wk3, const float* Wqm,
                         float* Wcat, float* WcatT) {
  int idx = blockIdx.x * blockDim.x + threadIdx.x;
  if (idx >= 128 * 640) return;
  int k = idx / 640, j = idx % 640;
  const float* Ws[5] = {Wq2, Wk2, Wq3, Wk3, Wqm};
  float w = Ws[j >> 7][k * 128 + (j & 127)];
  Wcat[k * 640 + j] = w;
  WcatT[j * 128 + k] = w;
}

// ---------- K4: Km = B_mem @ W_Km  (32x128) ----------
__global__ void k_km(const float* __restrict__ Bm, const float* __restrict__ Wkm,
                     float* Km) {
  int idx = blockIdx.x * blockDim.x + threadIdx.x;
  if (idx >= KM * DD) return;
  int k = idx >> 7, d = idx & 127;
  float s = 0.f;
  for (int j = 0; j < DD; ++j) s += Bm[k * DD + j] * Wkm[j * DD + d];
  Km[idx] = s;
}

// ---------- K5: a2 edge MLP (gelu exact) ----------
__global__ void k_a2(const float* __restrict__ ef, const float* __restrict__ W1,
                     const float* __restrict__ b1, const float* __restrict__ W2,
                     const float* __restrict__ b2p, float* a2) {
  int e = blockIdx.x * blockDim.x + threadIdx.x;
  if (e >= EE2) return;
  float f0 = ef[e * 5 + 0], f1 = ef[e * 5 + 1], f2 = ef[e * 5 + 2],
        f3 = ef[e * 5 + 3], f4 = ef[e * 5 + 4];
  float acc = 0.f;
  for (int h = 0; h < DD; ++h) {
    float t = b1[h] + f0 * W1[h] + f1 * W1[128 + h] + f2 * W1[256 + h] +
              f3 * W1[384 + h] + f4 * W1[512 + h];
    float g = 0.5f * t * (1.f + erff(t * 0.7071067811865476f));
    acc += g * W2[h];
  }
  a2[e] = acc + b2p[0];
}

// ---------- K6: Z = LayerNorm(X) @ Wcat   via V_WMMA_F32_16X16X4_F32 ----------
// One wave computes a 16(M) x 64(N) strip; 4 accumulators; K loop step 4.
__global__ void k_gemm_z(const float* __restrict__ X, const float* __restrict__ mu,
                         const float* __restrict__ inv, const float* __restrict__ gamma,
                         const float* __restrict__ beta, const float* __restrict__ Wcat,
                         float* __restrict__ Z) {
  const int MT = NN / 16, NT = 640 / 64;
  int wid = (blockIdx.x * blockDim.x + threadIdx.x) >> 5;  // wave-uniform
  if (wid >= MT * NT) return;                               // whole wave exits
  int lane = threadIdx.x & 31;
  bool hi = lane >= 16;
  int mt = wid / NT, nt = wid % NT;
  int r = mt * 16 + (lane & 15);
  float muv = mu[r], iv = inv[r];
  int col = nt * 64 + (lane & 15);
  v8f a0 = {}, a1 = {}, a2v = {}, a3 = {};
  for (int k = 0; k < DD; k += 4) {
    int k0 = k + (hi ? 2 : 0);
    v2f A;
    A.x = (X[r * DD + k0] - muv) * iv * gamma[k0] + beta[k0];
    A.y = (X[r * DD + k0 + 1] - muv) * iv * gamma[k0 + 1] + beta[k0 + 1];
    const float* B0 = Wcat + k0 * 640;
    const float* B1 = Wcat + (k0 + 1) * 640;
    v2f Bv0, Bv1, Bv2, Bv3;
    Bv0.x = B0[col];      Bv0.y = B1[col];
    Bv1.x = B0[col + 16]; Bv1.y = B1[col + 16];
    Bv2.x = B0[col + 32]; Bv2.y = B1[col + 32];
    Bv3.x = B0[col + 48]; Bv3.y = B1[col + 48];
    a0 = __builtin_amdgcn_wmma_f32_16x16x4_f32(false, A, false, Bv0, (short)0, a0, false, false);
    a1 = __builtin_amdgcn_wmma_f32_16x16x4_f32(false, A, false, Bv1, (short)0, a1, false, false);
    a2v = __builtin_amdgcn_wmma_f32_16x16x4_f32(false, A, false, Bv2, (short)0, a2v, false, false);
    a3 = __builtin_amdgcn_wmma_f32_16x16x4_f32(false, A, false, Bv3, (short)0, a3, false, false);
  }
  int rowbase = mt * 16 + (hi ? 8 : 0);
#pragma unroll
  for (int j = 0; j < 8; ++j) {
    float* zr = Z + (rowbase + j) * 640 + col;
    zr[0] = a0[j]; zr[16] = a1[j]; zr[32] = a2v[j]; zr[48] = a3[j];
  }
}

// ---------- K7: memory softmax fwd+bwd (Ems, dQm) ----------
__global__ void k_sm(const float* __restrict__ Z, const float* __restrict__ Km,
                     const float* sc, float* dZ, float* Ems_acc) {
  __shared__ float KmL[KM * 132];  // stride 132 avoids bank conflicts
  __shared__ float pbuf[8][32];
  for (int i = threadIdx.x; i < KM * DD; i += blockDim.x)
    KmL[(i >> 7) * 132 + (i & 127)] = Km[i];
  __syncthreads();
  int lane = threadIdx.x & 31, w = threadIdx.x >> 5;
  int node = blockIdx.x * 8 + w;
  if (node >= NN) return;
  float lamm = sc[2], bm = sc[5];
  const float* q = Z + (size_t)node * 640 + 512;
  const float* kr = KmL + lane * 132;
  float s = 0.f;
  for (int d = 0; d < DD; ++d) s += q[d] * kr[d];
  float t = bm * s * SCALE;
  float m = wave_max(t);
  float e = expf(t - m);
  float z = wave_sum(e);
  pbuf[w][lane] = e / z;
  if (lane == 0) atomAddF(Ems_acc, m + logf(z));
  float o0 = 0.f, o1 = 0.f, o2 = 0.f, o3 = 0.f;
  for (int k = 0; k < KM; ++k) {
    float pk = pbuf[w][k];
    const float* k2 = KmL + k * 132;
    o0 += pk * k2[lane];       o1 += pk * k2[lane + 32];
    o2 += pk * k2[lane + 64];  o3 += pk * k2[lane + 96];
  }
  float c = -lamm * SCALE;
  float* out = dZ + (size_t)node * 640 + 512;
  out[lane] = c * o0; out[lane + 32] = c * o1;
  out[lane + 64] = c * o2; out[lane + 96] = c * o3;
}

// ---------- K8: s2 scores + segment max (one wave per edge) ----------
__global__ void k_s2(const float* __restrict__ Z, const int* c2, const int* u2,
                     const float* a2, const float* sc, float* s2, unsigned* m2e) {
  int lane = threadIdx.x & 31;
  int e = blockIdx.x * 8 + (threadIdx.x >> 5);
  if (e >= EE2) return;
  int c = c2[e], u = u2[e];
  const float* Q = Z + (size_t)c * 640;
  const float* Kp = Z + (size_t)u * 640 + 128;
  float s = 0.f;
#pragma unroll
  for (int j = 0; j < 4; ++j) { int d = lane + 32 * j; s += Q[d] * Kp[d]; }
  s = wave_sum(s);
  if (lane == 0) {
    float sv = s * SCALE + a2[e];
    s2[e] = sv;
    atomicMax(&m2e[c], enc_f(sc[3] * sv));
  }
}

// ---------- K9: s3 scores + segment max ----------
__global__ void k_s3(const float* __restrict__ Z, const int* c3, const int* u3,
                     const int* v3, const int* tt, const float* __restrict__ T,
                     const float* sc, float* s3, unsigned* m3e) {
  int lane = threadIdx.x & 31;
  int e = blockIdx.x * 8 + (threadIdx.x >> 5);
  if (e >= EE3) return;
  int c = c3[e], u = u3[e], v = v3[e];
  const float* Q = Z + (size_t)c * 640 + 256;
  const float* Ku = Z + (size_t)u * 640 + 384;
  const float* Kv = Z + (size_t)v * 640 + 384;
  const float* Tp = T + tt[e] * DD;
  float s = 0.f;
#pragma unroll
  for (int j = 0; j < 4; ++j) {
    int d = lane + 32 * j;
    s += Q[d] * Ku[d] * Kv[d] * Tp[d];
  }
  s = wave_sum(s);
  if (lane == 0) {
    float sv = s * SCALE;
    s3[e] = sv;
    atomicMax(&m3e[c], enc_f(sc[4] * sv));
  }
}

// ---------- K10/K11: segment exp-sum accumulation ----------
__global__ void k_zacc2(const float* s2, const int* c2, const float* sc,
                        const unsigned* m2e, float* z2) {
  int e = blockIdx.x * blockDim.x + threadIdx.x;
  if (e >= EE2) return;
  int c = c2[e];
  atomAddF(&z2[c], expf(sc[3] * s2[e] - dec_f(m2e[c])));
}
__global__ void k_zacc3(const float* s3, const int* c3, const float* sc,
                        const unsigned* m3e, float* z3) {
  int e = blockIdx.x * blockDim.x + threadIdx.x;
  if (e >= EE3) return;
  int c = c3[e];
  atomAddF(&z3[c], expf(sc[4] * s3[e] - dec_f(m3e[c])));
}

// ---------- K12: sum of segment LSEs (E2s, E3s) ----------
__global__ void k_lse(const unsigned* m2e, const float* z2,
                      const unsigned* m3e, const float* z3, float* acc) {
  int i = blockIdx.x * blockDim.x + threadIdx.x;
  if (i >= NN) return;
  float m2 = dec_f(m2e[i]);
  if (z2[i] > 0.f && isfinite(m2))
    atomAddF(&acc[0], m2 + logf(fmaxf(z2[i], 1e-30f)));
  float m3 = dec_f(m3e[i]);
  if (z3[i] > 0.f && isfinite(m3))
    atomAddF(&acc[1], m3 + logf(fmaxf(z3[i], 1e-30f)));
}

// ---------- K13: backward scatter for pairwise edges ----------
__global__ void k_bwd2(const float* __restrict__ Z, float* __restrict__ dZ,
                       const int* c2, const int* u2, const float* s2,
                       const unsigned* m2e, const float* z2, const float* sc) {
  int lane = threadIdx.x & 31;
  int e = blockIdx.x * 8 + (threadIdx.x >> 5);
  if (e >= EE2) return;
  int c = c2[e], u = u2[e];
  float t = sc[3] * s2[e];
  float wgt = expf(t - dec_f(m2e[c])) / z2[c];
  float g = -sc[0] * wgt * SCALE;
  const float* Q = Z + (size_t)c * 640;
  const float* Kp = Z + (size_t)u * 640 + 128;
  float* dQ = dZ + (size_t)c * 640;
  float* dK = dZ + (size_t)u * 640 + 128;
#pragma unroll
  for (int j = 0; j < 4; ++j) {
    int d = lane + 32 * j;
    atomAddF(dQ + d, g * Kp[d]);
    atomAddF(dK + d, g * Q[d]);
  }
}

// ---------- K14: backward scatter for triple edges ----------
__global__ void k_bwd3(const float* __restrict__ Z, float* __restrict__ dZ,
                       const int* c3, const int* u3, const int* v3, const int* tt,
                       const float* __restrict__ T, const float* s3,
                       const unsigned* m3e, const float* z3, const float* sc) {
  int lane = threadIdx.x & 31;
  int e = blockIdx.x * 8 + (threadIdx.x >> 5);
  if (e >= EE3) return;
  int c = c3[e], u = u3[e], v = v3[e];
  float t = sc[4] * s3[e];
  float wgt = expf(t - dec_f(m3e[c])) / z3[c];
  float g = -sc[1] * wgt * SCALE;
  const float* Q = Z + (size_t)c * 640 + 256;
  const float* Ku = Z + (size_t)u * 640 + 384;
  const float* Kv = Z + (size_t)v * 640 + 384;
  const float* Tp = T + tt[e] * DD;
  float* dQ = dZ + (size_t)c * 640 + 256;
  float* dKu = dZ + (size_t)u * 640 + 384;
  float* dKv = dZ + (size_t)v * 640 + 384;
#pragma unroll
  for (int j = 0; j < 4; ++j) {
    int d = lane + 32 * j;
    float q = Q[d], ku = Ku[d], kv = Kv[d], tp = Tp[d];
    atomAddF(dQ + d, g * ku * kv * tp);
    atomAddF(dKu + d, g * q * kv * tp);
    atomAddF(dKv + d, g * q * ku * tp);
  }
}

// ---------- K15: dG = dZ @ WcatT   (N x 640)(640 x 128) via WMMA f32 ----------
__global__ void k_gemm_dg(const float* __restrict__ dZ, const float* __restrict__ WcatT,
                          float* __restrict__ dG) {
  const int MT = NN / 16, NT = 128 / 64;
  int wid = (blockIdx.x * blockDim.x + threadIdx.x) >> 5;
  if (wid >= MT * NT) return;
  int lane = threadIdx.x & 31;
  bool hi = lane >= 16;
  int mt = wid / NT, nt = wid % NT;
  int r = mt * 16 + (lane & 15);
  int col = nt * 64 + (lane & 15);
  v8f a0 = {}, a1 = {}, a2v = {}, a3 = {};
  for (int k = 0; k < 640; k += 4) {
    int k0 = k + (hi ? 2 : 0);
    v2f A;
    A.x = dZ[(size_t)r * 640 + k0];
    A.y = dZ[(size_t)r * 640 + k0 + 1];
    const float* B0 = WcatT + k0 * 128;
    const float* B1 = WcatT + (k0 + 1) * 128;
    v2f Bv0, Bv1, Bv2, Bv3;
    Bv0.x = B0[col];      Bv0.y = B1[col];
    Bv1.x = B0[col + 16]; Bv1.y = B1[col + 16];
    Bv2.x = B0[col + 32]; Bv2.y = B1[col + 32];
    Bv3.x = B0[col + 48]; Bv3.y = B1[col + 48];
    a0 = __builtin_amdgcn_wmma_f32_16x16x4_f32(false, A, false, Bv0, (short)0, a0, false, false);
    a1 = __builtin_amdgcn_wmma_f32_16x16x4_f32(false, A, false, Bv1, (short)0, a1, false, false);
    a2v = __builtin_amdgcn_wmma_f32_16x16x4_f32(false, A, false, Bv2, (short)0, a2v, false, false);
    a3 = __builtin_amdgcn_wmma_f32_16x16x4_f32(false, A, false, Bv3, (short)0, a3, false, false);
  }
  int rowbase = mt * 16 + (hi ? 8 : 0);
#pragma unroll
  for (int j = 0; j < 8; ++j) {
    float* gr = dG + (size_t)(rowbase + j) * DD + col;
    gr[0] = a0[j]; gr[16] = a1[j]; gr[32] = a2v[j]; gr[48] = a3[j];
  }
}

// ---------- K16: LayerNorm backward + clip + update ----------
__global__ void k_ln_bwd(const float* __restrict__ X, const float* __restrict__ dG,
                         const float* __restrict__ mu, const float* __restrict__ inv,
                         const float* __restrict__ gamma, const float* sc,
                         float* __restrict__ out) {
  int lane = threadIdx.x & 31;
  int r = blockIdx.x * 8 + (threadIdx.x >> 5);
  if (r >= NN) return;
  float muv = mu[r], iv = inv[r];
  float step = sc[6] * DAMPC;
  float xh[4], dxh[4], xv[4];
  float s1 = 0.f, s2 = 0.f;
#pragma unroll
  for (int j = 0; j < 4; ++j) {
    int d = lane + 32 * j;
    xv[j] = X[r * DD + d];
    xh[j] = (xv[j] - muv) * iv;
    dxh[j] = dG[(size_t)r * DD + d] * gamma[d];
    s1 += dxh[j];
    s2 += dxh[j] * xh[j];
  }
  s1 = wave_sum(s1) * (1.f / DD);
  s2 = wave_sum(s2) * (1.f / DD);
  float g[4], gn = 0.f;
#pragma unroll
  for (int j = 0; j < 4; ++j) {
    g[j] = iv * (dxh[j] - s1 - xh[j] * s2);
    gn += g[j] * g[j];
  }
  gn = sqrtf(wave_sum(gn));
  float cf = fminf(1.f, GRAD_CLIPC / fmaxf(gn, 1e-6f));
  float xn[4], sn = 0.f;
#pragma unroll
  for (int j = 0; j < 4; ++j) {
    xn[j] = xv[j] - step * cf * g[j];
    sn += xn[j] * xn[j];
  }
  sn = sqrtf(wave_sum(sn));
  float sf = fminf(1.f, STATE_CLIPC / fmaxf(sn, 1e-6f));
#pragma unroll
  for (int j = 0; j < 4; ++j) out[r * DD + lane + 32 * j] = xn[j] * sf;
}

// ---------- K17: energy finalize ----------
__global__ void k_final(const float* sc, const float* acc, float* out) {
  if (threadIdx.x == 0 && blockIdx.x == 0)
    out[NN * DD] = -(sc[0] / sc[3]) * acc[0] - (sc[1] / sc[4]) * acc[1] -
                   (sc[2] / sc[5]) * acc[2];
}

extern "C" void kernel_launch(void* const* d_in, const int* in_sizes, int n_in,
                              void* d_out, int out_size, void* d_ws, size_t ws_size,
                              hipStream_t stream) {
  const float* X = (const float*)d_in[0];
  const int *c2 = (const int*)d_in[1], *u2 = (const int*)d_in[2];
  const int *c3 = (const int*)d_in[3], *u3 = (const int*)d_in[4],
            *v3 = (const int*)d_in[5], *tt = (const int*)d_in[6];
  const float* ef = (const float*)d_in[7];
  const float *gamma = (const float*)d_in[8], *beta = (const float*)d_in[9];
  const float *Wq2 = (const float*)d_in[10], *Wk2 = (const float*)d_in[11];
  const float *Wq3 = (const float*)d_in[12], *Wk3 = (const float*)d_in[13];
  const float* T = (const float*)d_in[14];
  const float *Wqm = (const float*)d_in[15], *Wkm = (const float*)d_in[16];
  const float* Bm = (const float*)d_in[17];
  const float *W1 = (const float*)d_in[18], *b1 = (const float*)d_in[19];
  const float *W2 = (const float*)d_in[20], *b2p = (const float*)d_in[21];
  const float *l2r = (const float*)d_in[22], *l3r = (const float*)d_in[23],
              *lmr = (const float*)d_in[24];
  const float *b2r = (const float*)d_in[25], *b3r = (const float*)d_in[26],
              *bmr = (const float*)d_in[27];
  const float* ss = (const float*)d_in[28];
  float* out = (float*)d_out;

  // workspace layout (floats)
  float* w = (float*)d_ws;
  size_t o = 0;
  float* sc = w + o;       o += 8;
  float* acc = w + o;      o += 8;
  float* mu = w + o;       o += NN;
  float* inv = w + o;      o += NN;
  unsigned* m2e = (unsigned*)(w + o); o += NN;
  float* z2 = w + o;       o += NN;
  unsigned* m3e = (unsigned*)(w + o); o += NN;
  float* z3 = w + o;       o += NN;
  float* s2 = w + o;       o += EE2;
  float* a2 = w + o;       o += EE2;
  float* s3 = w + o;       o += EE3;
  float* Wcat = w + o;     o += 128 * 640;
  float* WcatT = w + o;    o += 640 * 128;
  float* Km = w + o;       o += KM * DD;
  float* Z = w + o;        o += (size_t)NN * 640;
  float* dZ = w + o;       o += (size_t)NN * 640;
  float* dG = Z;  // Z is dead after the backward scatters; reuse it for dG

  k_scalars<<<1, 64, 0, stream>>>(l2r, l3r, lmr, b2r, b3r, bmr, ss, sc, acc);
  hipMemsetAsync(dZ, 0, (size_t)NN * 640 * sizeof(float), stream);
  k_seg_init<<<(NN + 255) / 256, 256, 0, stream>>>(m2e, z2, m3e, z3);
  k_ln_stats<<<NN / 8, 256, 0, stream>>>(X, mu, inv);
  k_pack_w<<<(128 * 640) / 256, 256, 0, stream>>>(Wq2, Wk2, Wq3, Wk3, Wqm, Wcat, WcatT);
  k_km<<<(KM * DD) / 256, 256, 0, stream>>>(Bm, Wkm, Km);
  k_a2<<<(EE2 + 255) / 256, 256, 0, stream>>>(ef, W1, b1, W2, b2p, a2);

  // forward projection GEMM (WMMA f32): 3125*10 waves, 8 waves/block
  k_gemm_z<<<(3125 * 10 + 7) / 8, 256, 0, stream>>>(X, mu, inv, gamma, beta, Wcat, Z);

  k_sm<<<NN / 8, 256, 0, stream>>>(Z, Km, sc, dZ, acc + 2);
  k_s2<<<EE2 / 8, 256, 0, stream>>>(Z, c2, u2, a2, sc, s2, m2e);
  k_s3<<<EE3 / 8, 256, 0, stream>>>(Z, c3, u3, v3, tt, T, sc, s3, m3e);
  k_zacc2<<<(EE2 + 255) / 256, 256, 0, stream>>>(s2, c2, sc, m2e, z2);
  k_zacc3<<<(EE3 + 255) / 256, 256, 0, stream>>>(s3, c3, sc, m3e, z3);
  k_lse<<<(NN + 255) / 256, 256, 0, stream>>>(m2e, z2, m3e, z3, acc);
  k_bwd2<<<EE2 / 8, 256, 0, stream>>>(Z, dZ, c2, u2, s2, m2e, z2, sc);
  k_bwd3<<<EE3 / 8, 256, 0, stream>>>(Z, dZ, c3, u3, v3, tt, T, s3, m3e, z3, sc);

  // backward projection GEMM (WMMA f32): 3125*2 waves
  k_gemm_dg<<<(3125 * 2 + 7) / 8, 256, 0, stream>>>(dZ, WcatT, dG);

  k_ln_bwd<<<NN / 8, 256, 0, stream>>>(X, dG, mu, inv, gamma, sc, out);
  k_final<<<1, 64, 0, stream>>>(sc, acc, out);
}